// CATS_SwiGLU_17772574671210
// MI455X (gfx1250) — compile-verified
//
#include <hip/hip_runtime.h>
#include <hip/hip_bf16.h>

#define D_MODEL 4096
#define D_FF    11008
#define S1      4      // phase-1 split-K factor
#define S3      4      // phase-3 split-K factor
#define PAD     16

typedef __attribute__((ext_vector_type(2))) float v2f;
typedef __attribute__((ext_vector_type(8))) float v8f;

__device__ __forceinline__ v8f wmma_f32_4(v2f a, v2f b, v8f c) {
    return __builtin_amdgcn_wmma_f32_16x16x4_f32(
        /*neg_a=*/false, a, /*neg_b=*/false, b,
        /*c_mod=*/(short)0, c, /*reuse_a=*/false, /*reuse_b=*/false);
}

// ---------------------------------------------------------------------------
// Phase 1: part1[s][j] = partial of (x @ Wgatet)[j] over K-slice s.
// One wave per 16-column tile; A row0 = x chunk (lanes 0/16 carry it, other
// lanes zeroed via mask-multiply -- no EXEC branching), B = 4x16 Wgatet tile.
// grid = (688/8, S1), block = 256 (8 waves)
// ---------------------------------------------------------------------------
__global__ __launch_bounds__(256) void k1_gate(const float* __restrict__ x,
                                               const float* __restrict__ Wg,
                                               float* __restrict__ part1) {
    const int wid  = threadIdx.x >> 5;
    const int lane = threadIdx.x & 31;
    const int t    = blockIdx.x * 8 + wid;          // 0..687 tile id
    const int n0   = t * 16;
    const int n    = lane & 15;
    const int c    = (lane < 16) ? 0 : 2;
    const float am = (n == 0) ? 1.f : 0.f;          // A-row-0 mask
    const int kbase = blockIdx.y * (D_MODEL / S1);  // 1024-deep slice

    const float* bp = Wg + (size_t)(kbase + c) * D_FF + n0 + n;  // B: row kbase+c, col n0+n
    const float* xp = x + kbase + c;

    v8f acc0 = {}; v8f acc1 = {};
    #pragma unroll 2
    for (int k = 0; k < D_MODEL / S1; k += 8) {
        v2f xv0 = *(const v2f*)(xp + k);            // broadcast load, all lanes
        v2f a0; a0.x = xv0.x * am; a0.y = xv0.y * am;
        v2f b0; b0.x = bp[(size_t)k * D_FF];
                b0.y = bp[(size_t)k * D_FF + D_FF];
        acc0 = wmma_f32_4(a0, b0, acc0);

        v2f xv1 = *(const v2f*)(xp + k + 4);
        v2f a1; a1.x = xv1.x * am; a1.y = xv1.y * am;
        v2f b1; b1.x = bp[(size_t)(k + 4) * D_FF];
                b1.y = bp[(size_t)(k + 4) * D_FF + D_FF];
        acc1 = wmma_f32_4(a1, b1, acc1);
    }
    // D row 0 (M=0) lives in acc[0] on lanes 0..15 (N = lane)
    if (lane < 16) part1[(size_t)blockIdx.y * D_FF + n0 + lane] = acc0[0] + acc1[0];
}

// ---------------------------------------------------------------------------
// Phase 1b: x1[j] = silu(sum_s part1[s][j])     grid = 43x256
// ---------------------------------------------------------------------------
__global__ __launch_bounds__(256) void k2_silu(const float* __restrict__ part1,
                                               float* __restrict__ x1) {
    int j = blockIdx.x * 256 + threadIdx.x;
    float v = 0.f;
    #pragma unroll
    for (int s = 0; s < S1; ++s) v += part1[(size_t)s * D_FF + j];
    x1[j] = v / (1.f + __expf(-v));
}

// ---------------------------------------------------------------------------
// Compaction: single ordered wave builds flagged-index list (deterministic).
// grid = 1 block, 32 threads.
// ---------------------------------------------------------------------------
__global__ __launch_bounds__(32) void k3_compact(const float* __restrict__ x1,
                                                 const float* __restrict__ thr_p,
                                                 int* __restrict__ idxc,
                                                 float* __restrict__ x1c,
                                                 int* __restrict__ meta) {
    const int lane = threadIdx.x;
    const float thr = thr_p[0];
    int base = 0;
    const unsigned below = (1u << lane) - 1u;
    for (int chunk = 0; chunk < (D_FF + 31) / 32; ++chunk) {
        int j = chunk * 32 + lane;
        bool f = (j < D_FF) && (fabsf(x1[j]) > thr);
        unsigned mask = (unsigned)__ballot(f);
        if (f) {
            int pos = base + __popc(mask & below);
            idxc[pos] = j;
            x1c[pos]  = x1[j];
        }
        base += __popc(mask);
    }
    const int m   = base;
    const int r16 = (m + 15) & ~15;
    for (int i = m + lane; i < r16; i += 32) { idxc[i] = 0; x1c[i] = 0.f; }
    if (lane == 0) { meta[0] = m; meta[1] = r16; }
}

// ---------------------------------------------------------------------------
// Phase 2: gathered up-proj. One wave per 16 flagged rows:
// A = 16 gathered Wup rows (K=4 chunk), B = broadcast x chunk.
// D[m][*] = dot(Wup[idx_m], x); zc[i] = dot_i * x1c[i].
// grid = 86, block = 256; waves beyond r16/16 exit.
// ---------------------------------------------------------------------------
__global__ __launch_bounds__(256) void k4_up(const float* __restrict__ x,
                                             const float* __restrict__ Wup,
                                             const int*   __restrict__ idxc,
                                             const float* __restrict__ x1c,
                                             const int*   __restrict__ meta,
                                             float* __restrict__ zc) {
    const int wid  = threadIdx.x >> 5;
    const int lane = threadIdx.x & 31;
    const int g    = blockIdx.x * 8 + wid;
    const int r16  = meta[1];
    if (g * 16 >= r16) return;

    const int c   = (lane < 16) ? 0 : 2;
    const int row = idxc[g * 16 + (lane & 15)];       // M = lane&15
    const float* ap = Wup + (size_t)row * D_MODEL + c;
    const float* xp = x + c;

    v8f acc0 = {}; v8f acc1 = {};
    #pragma unroll 2
    for (int k = 0; k < D_MODEL; k += 8) {
        v2f a0 = *(const v2f*)(ap + k);
        v2f b0 = *(const v2f*)(xp + k);
        acc0 = wmma_f32_4(a0, b0, acc0);
        v2f a1 = *(const v2f*)(ap + k + 4);
        v2f b1 = *(const v2f*)(xp + k + 4);
        acc1 = wmma_f32_4(a1, b1, acc1);
    }
    // D replicated across N: lane 0 holds rows 0..7 in acc[0..7] (N=0),
    // lane 16 holds rows 8..15.
    if (lane == 0) {
        #pragma unroll
        for (int i = 0; i < 8; ++i)
            zc[g * 16 + i] = (acc0[i] + acc1[i]) * x1c[g * 16 + i];
    } else if (lane == 16) {
        #pragma unroll
        for (int i = 0; i < 8; ++i)
            zc[g * 16 + 8 + i] = (acc0[i] + acc1[i]) * x1c[g * 16 + 8 + i];
    }
}

// ---------------------------------------------------------------------------
// Phase 3: out = zc @ gathered Wdownt rows. One wave per 16 output cols,
// split-K over the flagged list. A row0 = zc chunk (mask-multiply select),
// B = gathered 4x16 Wdownt tile.
// grid = (32, S3), block = 256.
// ---------------------------------------------------------------------------
__global__ __launch_bounds__(256) void k5_down(const float* __restrict__ Wd,
                                               const int*   __restrict__ idxc,
                                               const float* __restrict__ zc,
                                               const int*   __restrict__ meta,
                                               float* __restrict__ part3) {
    const int wid  = threadIdx.x >> 5;
    const int lane = threadIdx.x & 31;
    const int t    = blockIdx.x * 8 + wid;            // 0..255 column tile
    const int n0   = t * 16;
    const int n    = lane & 15;
    const int c    = (lane < 16) ? 0 : 2;
    const float am = (n == 0) ? 1.f : 0.f;

    const int r16    = meta[1];
    const int total4 = r16 >> 2;                      // # of 4-row steps
    const int per    = (total4 + S3 - 1) / S3;
    const int k4s    = blockIdx.y * per;
    const int k4e    = min(k4s + per, total4);

    v8f acc = {};
    #pragma unroll 2
    for (int k4 = k4s; k4 < k4e; ++k4) {
        const int k0 = k4 * 4;
        const int r0 = idxc[k0 + c];
        const int r1 = idxc[k0 + c + 1];
        v2f zv = *(const v2f*)(zc + k0 + c);          // broadcast load, all lanes
        v2f a; a.x = zv.x * am; a.y = zv.y * am;
        v2f b; b.x = Wd[(size_t)r0 * D_MODEL + n0 + n];
               b.y = Wd[(size_t)r1 * D_MODEL + n0 + n];
        acc = wmma_f32_4(a, b, acc);
    }
    if (lane < 16) part3[(size_t)blockIdx.y * D_MODEL + n0 + lane] = acc[0];
}

// ---------------------------------------------------------------------------
// Phase 3b: out[k] = sum_s part3[s][k]     grid = 16x256
// ---------------------------------------------------------------------------
__global__ __launch_bounds__(256) void k6_reduce(const float* __restrict__ part3,
                                                 float* __restrict__ out) {
    int k = blockIdx.x * 256 + threadIdx.x;
    float v = 0.f;
    #pragma unroll
    for (int s = 0; s < S3; ++s) v += part3[(size_t)s * D_MODEL + k];
    out[k] = v;
}

extern "C" void kernel_launch(void* const* d_in, const int* in_sizes, int n_in,
                              void* d_out, int out_size, void* d_ws, size_t ws_size,
                              hipStream_t stream) {
    const float* x   = (const float*)d_in[0];  // [1,1,4096]
    const float* Wup = (const float*)d_in[1];  // [11008,4096]
    const float* Wg  = (const float*)d_in[2];  // [4096,11008]
    const float* Wd  = (const float*)d_in[3];  // [11008,4096]
    const float* thr = (const float*)d_in[4];  // scalar
    float* out = (float*)d_out;                // [4096]

    // workspace carve-up
    float* part1 = (float*)d_ws;                       // S1 * D_FF
    float* x1    = part1 + (size_t)S1 * D_FF;          // D_FF
    int*   idxc  = (int*)(x1 + D_FF);                  // D_FF + PAD
    float* x1c   = (float*)(idxc + (D_FF + PAD));      // D_FF + PAD
    float* zc    = x1c + (D_FF + PAD);                 // D_FF + PAD
    float* part3 = zc + (D_FF + PAD);                  // S3 * D_MODEL
    int*   meta  = (int*)(part3 + (size_t)S3 * D_MODEL);

    // 1) gate GEMV (dense, split-K) -> partials
    k1_gate<<<dim3((D_FF / 16) / 8, S1), 256, 0, stream>>>(x, Wg, part1);
    // 2) reduce + silu
    k2_silu<<<D_FF / 256, 256, 0, stream>>>(part1, x1);
    // 3) deterministic flag compaction (single wave)
    k3_compact<<<1, 32, 0, stream>>>(x1, thr, idxc, x1c, meta);
    // 4) gathered up-proj over flagged rows only
    k4_up<<<(D_FF / 16) / 8, 256, 0, stream>>>(x, Wup, idxc, x1c, meta, zc);
    // 5) gathered down-proj (split-K over flag list) -> partials
    k5_down<<<dim3((D_MODEL / 16) / 8, S3), 256, 0, stream>>>(Wd, idxc, zc, meta, part3);
    // 6) final reduce
    k6_reduce<<<D_MODEL / 256, 256, 0, stream>>>(part3, out);
}